// HierarchicalAttention_37331855737398
// MI455X (gfx1250) — compile-verified
//
#include <hip/hip_runtime.h>
#include <hip/hip_bf16.h>
#include <math.h>

// ---------------------------------------------------------------------------
// HierarchicalAttention on MI455X (gfx1250, wave32, WMMA)
// - fp32 reference -> V_WMMA_F32_16X16X4_F32 for all GEMMs.
// - Attention matrix [8,8,1024,1024] computed directly in d_out
//   (scores -> in-place softmax -> re-read for ctx): ~0.5 GB HBM traffic,
//   ~23 us floor at 23.3 TB/s; everything else stays in the 192 MB L2.
// - Generic GEMM: 4-wave blocks stage the shared 16x256 weight tile into LDS
//   with GLOBAL_LOAD_ASYNC_TO_LDS_B128 (ASYNCcnt) and read B-fragments via
//   ds_load; A streams from global (L2-hot).
// - Scores: each wave computes 16x64 (4 accumulators, A-frag reused 4x).
//   ctx: each wave computes 16x32 (2 accumulators).
// - LSTM recurrence: persistent workgroup per direction, h/c/gates in LDS,
//   per-step h@Whh.T via WMMA with batch M=8 padded to 16.
// ---------------------------------------------------------------------------

typedef __attribute__((ext_vector_type(2))) float v2f;
typedef __attribute__((ext_vector_type(8))) float v8f;

__device__ __forceinline__ v8f wmma4(v2f a, v2f b, v8f c) {
  return __builtin_amdgcn_wmma_f32_16x16x4_f32(false, a, false, b, (short)0, c,
                                               false, false);
}

__device__ __forceinline__ float sigmf(float x) { return 1.0f / (1.0f + expf(-x)); }

// Async-stage 16 bytes global -> LDS (no VGPR round trip, ASYNCcnt-tracked).
__device__ __forceinline__ void async_g2l_b128(unsigned lds_byte_off,
                                               const float* gaddr) {
  asm volatile("global_load_async_to_lds_b128 %0, %1, off"
               :: "v"(lds_byte_off), "v"(gaddr) : "memory");
}
__device__ __forceinline__ void wait_async0() {
  asm volatile("s_wait_asynccnt 0" ::: "memory");
}

// ---------------------------------------------------------------------------
// Generic fp32 WMMA GEMM for D = A @ W.T (+bias(+bias2))(+add)
// A: [M x 256] (lda = 256), W: [N x 256] row-major torch weight.
// Block = 128 threads (4 waves) computing a 64x16 tile; the 16x256 W tile is
// shared by all 4 waves and staged in LDS via async global->LDS loads.
// grid = (M/64, N/16)
// ---------------------------------------------------------------------------
__global__ void __launch_bounds__(128)
gemm_wmma_f32(const float* __restrict__ A, int lda,
              const float* __restrict__ W,
              const float* __restrict__ bias, const float* __restrict__ bias2,
              const float* __restrict__ addres,
              float* __restrict__ Dst, int ldd) {
  __shared__ float bs[16 * 256];
  const int tid = threadIdx.x;
  const int n0 = blockIdx.y * 16;
  // --- stage W[n0..n0+15][0..255] (16 KB) into LDS, 8 x b128 per thread ---
#pragma unroll
  for (int i = 0; i < 8; ++i) {
    const int idx = tid + i * 128;        // 0..1023 b128 chunks
    const int row = idx >> 6;             // 0..15
    const int c4 = (idx & 63) << 2;       // float col, multiple of 4
    const float* gp = W + (size_t)(n0 + row) * 256 + c4;
    const unsigned la = (unsigned)(size_t)&bs[row * 256 + c4];
    async_g2l_b128(la, gp);
  }
  wait_async0();
  __syncthreads();

  const int lane = tid & 31, half = lane >> 4, r = lane & 15;
  const int m0 = blockIdx.x * 64 + (tid >> 5) * 16;
  const float* arow = A + (size_t)(m0 + r) * lda;
  const float* bsr = &bs[r * 256];
  v8f c = {};
  for (int k0 = 0; k0 < 256; k0 += 4) {
    const int ka = k0 + 2 * half;
    v2f av, bv;
    av[0] = arow[ka]; av[1] = arow[ka + 1];
    bv[0] = bsr[ka];  bv[1] = bsr[ka + 1];
    c = wmma4(av, bv, c);
  }
  float bn = 0.0f;
  if (bias)  bn += bias[n0 + r];
  if (bias2) bn += bias2[n0 + r];
#pragma unroll
  for (int i = 0; i < 8; ++i) {
    const int m = m0 + i + 8 * half;
    float val = c[i] + bn;
    if (addres) val += addres[(size_t)m * ldd + n0 + r];
    Dst[(size_t)m * ldd + n0 + r] = val;
  }
}

// ---------------------------------------------------------------------------
// Attention scores: attn[bh,q,k'] = scale * dot(Q[b,q,h,:], K[b,k',h,:]).
// One wave per 16x64 output strip (4 accumulators, A-frag reused 4x), K=32.
// grid = (64 q-tiles, 16 n-strips, 64 bh)
// ---------------------------------------------------------------------------
__global__ void __launch_bounds__(32)
attn_scores_wmma(const float* __restrict__ Qp, const float* __restrict__ Kp,
                 float* __restrict__ attn, float scale) {
  const int bh = blockIdx.z;
  const int b = bh >> 3, h = bh & 7;
  const int lane = threadIdx.x & 31, half = lane >> 4, r = lane & 15;
  const int m0 = blockIdx.x * 16, n0 = blockIdx.y * 64;
  const float* arow = Qp + ((size_t)(b * 1024 + m0 + r)) * 256 + h * 32;
  const float* b0r = Kp + ((size_t)(b * 1024 + n0 + r)) * 256 + h * 32;
  const float* b1r = b0r + 16 * 256;
  const float* b2r = b0r + 32 * 256;
  const float* b3r = b0r + 48 * 256;
  v8f c0 = {}, c1 = {}, c2 = {}, c3 = {};
#pragma unroll
  for (int k0 = 0; k0 < 32; k0 += 4) {
    const int ka = k0 + 2 * half;
    v2f av, bv0, bv1, bv2, bv3;
    av[0]  = arow[ka]; av[1]  = arow[ka + 1];
    bv0[0] = b0r[ka];  bv0[1] = b0r[ka + 1];
    bv1[0] = b1r[ka];  bv1[1] = b1r[ka + 1];
    bv2[0] = b2r[ka];  bv2[1] = b2r[ka + 1];
    bv3[0] = b3r[ka];  bv3[1] = b3r[ka + 1];
    c0 = wmma4(av, bv0, c0);
    c1 = wmma4(av, bv1, c1);
    c2 = wmma4(av, bv2, c2);
    c3 = wmma4(av, bv3, c3);
  }
  float* dbase = attn + (size_t)bh * 1024 * 1024;
#pragma unroll
  for (int i = 0; i < 8; ++i) {
    float* drow = dbase + (size_t)(m0 + i + 8 * half) * 1024 + r;
    drow[n0]      = c0[i] * scale;
    drow[n0 + 16] = c1[i] * scale;
    drow[n0 + 32] = c2[i] * scale;
    drow[n0 + 48] = c3[i] * scale;
  }
}

// ---------------------------------------------------------------------------
// In-place softmax over rows of length 1024. One 256-thread block per row.
// ---------------------------------------------------------------------------
__global__ void __launch_bounds__(256)
softmax1024(float* __restrict__ attn) {
  __shared__ float red[256];
  float* p = attn + (size_t)blockIdx.x * 1024;
  const int t = threadIdx.x;
  float x0 = p[t], x1 = p[t + 256], x2 = p[t + 512], x3 = p[t + 768];
  float mx = fmaxf(fmaxf(x0, x1), fmaxf(x2, x3));
  red[t] = mx; __syncthreads();
  for (int s = 128; s > 0; s >>= 1) { if (t < s) red[t] = fmaxf(red[t], red[t + s]); __syncthreads(); }
  mx = red[0]; __syncthreads();
  x0 = expf(x0 - mx); x1 = expf(x1 - mx); x2 = expf(x2 - mx); x3 = expf(x3 - mx);
  red[t] = x0 + x1 + x2 + x3; __syncthreads();
  for (int s = 128; s > 0; s >>= 1) { if (t < s) red[t] += red[t + s]; __syncthreads(); }
  const float inv = 1.0f / red[0];
  p[t] = x0 * inv; p[t + 256] = x1 * inv; p[t + 512] = x2 * inv; p[t + 768] = x3 * inv;
}

// ---------------------------------------------------------------------------
// ctx[b,q,h*32+n] = sum_k attn[bh,q,k] * V[b,k,h*32+n]
// One wave per 16(M)x32(N) strip (whole head, 2 accumulators), K = 1024.
// grid = (64 q-tiles, 64 bh)
// ---------------------------------------------------------------------------
__global__ void __launch_bounds__(32)
attn_ctx_wmma(const float* __restrict__ attn, const float* __restrict__ Vp,
              float* __restrict__ ctx) {
  const int bh = blockIdx.y;
  const int b = bh >> 3, h = bh & 7;
  const int lane = threadIdx.x & 31, half = lane >> 4, r = lane & 15;
  const int m0 = blockIdx.x * 16;
  const float* arow = attn + (size_t)bh * 1024 * 1024 + (size_t)(m0 + r) * 1024;
  const float* vb = Vp + (size_t)b * 1024 * 256 + h * 32 + r;
  v8f c0 = {}, c1 = {};
  for (int k0 = 0; k0 < 1024; k0 += 4) {
    const int ka = k0 + 2 * half;
    v2f av, bv0, bv1;
    av[0] = arow[ka]; av[1] = arow[ka + 1];
    const float* v0 = vb + (size_t)ka * 256;
    bv0[0] = v0[0];        bv0[1] = v0[256];
    bv1[0] = v0[16];       bv1[1] = v0[256 + 16];
    c0 = wmma4(av, bv0, c0);
    c1 = wmma4(av, bv1, c1);
  }
#pragma unroll
  for (int i = 0; i < 8; ++i) {
    float* crow = ctx + (size_t)(b * 1024 + m0 + i + 8 * half) * 256 + h * 32 + r;
    crow[0]  = c0[i];
    crow[16] = c1[i];
  }
}

// ---------------------------------------------------------------------------
// In-place LayerNorm over rows of 256 (input already holds out + residual).
// ---------------------------------------------------------------------------
__global__ void __launch_bounds__(256)
layernorm256(float* __restrict__ wa, const float* __restrict__ g,
             const float* __restrict__ bb) {
  __shared__ float red[256];
  float* p = wa + (size_t)blockIdx.x * 256;
  const int t = threadIdx.x;
  const float v = p[t];
  red[t] = v; __syncthreads();
  for (int s = 128; s > 0; s >>= 1) { if (t < s) red[t] += red[t + s]; __syncthreads(); }
  const float mean = red[0] * (1.0f / 256.0f); __syncthreads();
  const float d = v - mean;
  red[t] = d * d; __syncthreads();
  for (int s = 128; s > 0; s >>= 1) { if (t < s) red[t] += red[t + s]; __syncthreads(); }
  const float var = red[0] * (1.0f / 256.0f);
  p[t] = d * rsqrtf(var + 1e-5f) * g[t] + bb[t];
}

// ---------------------------------------------------------------------------
// Ragged per-sentence mean pooling. Block per (b,s), 256 threads (one per d).
// ---------------------------------------------------------------------------
__global__ void __launch_bounds__(256)
pool_mean(const float* __restrict__ wa, const int* __restrict__ sb,
          float* __restrict__ pooled) {
  const int bs = blockIdx.x;            // b*128 + s
  const int b = bs >> 7;
  const int start = sb[bs * 2], end = sb[bs * 2 + 1];
  const int t = threadIdx.x;
  float acc = 0.0f;
  for (int w = start; w < end; ++w)
    acc += wa[((size_t)b * 1024 + w) * 256 + t];
  const float cnt = fmaxf((float)(end - start), 1.0f);
  pooled[(size_t)bs * 256 + t] = acc / cnt;
}

// ---------------------------------------------------------------------------
// LSTM recurrence. gridDim.x = 2 (dir 0 = fwd, 1 = bwd), 512 threads/block.
// xW = precomputed x@Wih.T + bih + bhh, layout [dir][B*S, 512].
// ---------------------------------------------------------------------------
__global__ void __launch_bounds__(512)
lstm_rec(const float* __restrict__ xW, const float* __restrict__ whh_f,
         const float* __restrict__ whh_b, float* __restrict__ xnext) {
  const int dir = blockIdx.x;
  const float* whh = dir ? whh_b : whh_f;
  const float* xw = xW + (size_t)dir * 8 * 128 * 512;
  __shared__ float hb[8 * 128];
  __shared__ float cb[8 * 128];
  __shared__ float gb[8 * 512];
  const int tid = threadIdx.x;
  for (int p = tid; p < 1024; p += 512) { hb[p] = 0.0f; cb[p] = 0.0f; }
  __syncthreads();
  const int lane = tid & 31, half = lane >> 4, r = lane & 15;
  const int wave = tid >> 5;                    // 0..15
  const float amask = (r < 8) ? 1.0f : 0.0f;    // pad batch rows 8..15 with 0
  const int hrow = (r & 7) * 128;
  for (int step = 0; step < 128; ++step) {
    const int t = dir ? (127 - step) : step;
    for (int tt = 0; tt < 2; ++tt) {
      const int n0 = wave * 32 + tt * 16;
      const float* brow = whh + (size_t)(n0 + r) * 128;
      v8f c = {};
      for (int k0 = 0; k0 < 128; k0 += 4) {
        const int ka = k0 + 2 * half;
        v2f av, bv;
        av[0] = hb[hrow + ka] * amask;
        av[1] = hb[hrow + ka + 1] * amask;
        bv[0] = brow[ka]; bv[1] = brow[ka + 1];
        c = wmma4(av, bv, c);
      }
#pragma unroll
      for (int i = 0; i < 8; ++i) {
        const int mb = i + 8 * half;
        if (mb < 8)
          gb[mb * 512 + n0 + r] =
              c[i] + xw[((size_t)(mb * 128 + t)) * 512 + n0 + r];
      }
    }
    __syncthreads();
    for (int p = tid; p < 1024; p += 512) {
      const int b = p >> 7, j = p & 127;
      const float gi = gb[b * 512 + j];
      const float gf = gb[b * 512 + 128 + j];
      const float gg = gb[b * 512 + 256 + j];
      const float go = gb[b * 512 + 384 + j];
      const float cc = sigmf(gf) * cb[p] + sigmf(gi) * tanhf(gg);
      const float hh = sigmf(go) * tanhf(cc);
      cb[p] = cc; hb[p] = hh;
      xnext[((size_t)(b * 128 + t)) * 256 + dir * 128 + j] = hh;
    }
    __syncthreads();
  }
}

// ---------------------------------------------------------------------------
// Sentence scores: sc[b,s] = tanh(x[b,s]@a1w.T + a1b) @ a2w.T + a2b
// ---------------------------------------------------------------------------
__global__ void __launch_bounds__(128)
sent_score(const float* __restrict__ x, const float* __restrict__ a1w,
           const float* __restrict__ a1b, const float* __restrict__ a2w,
           const float* __restrict__ a2b, float* __restrict__ sc) {
  __shared__ float red[128];
  const int bs = blockIdx.x;
  const int t = threadIdx.x;
  const float* xr = x + (size_t)bs * 256;
  const float* wr = a1w + (size_t)t * 256;
  float acc = a1b[t];
  for (int k = 0; k < 256; ++k) acc += xr[k] * wr[k];
  red[t] = tanhf(acc) * a2w[t];
  __syncthreads();
  for (int s = 64; s > 0; s >>= 1) { if (t < s) red[t] += red[t + s]; __syncthreads(); }
  if (t == 0) sc[bs] = red[0] + a2b[0];
}

// ---------------------------------------------------------------------------
// Final doc vector + sentence softmax. Block per batch b, 256 threads.
// out[0:2048] = doc [8,256]; out[2048:3072] = sw [8,128].
// ---------------------------------------------------------------------------
__global__ void __launch_bounds__(256)
doc_reduce(const float* __restrict__ sc, const float* __restrict__ x,
           float* __restrict__ out) {
  __shared__ float red[128];
  __shared__ float sw[128];
  const int b = blockIdx.x;
  const int t = threadIdx.x;
  float v = (t < 128) ? sc[b * 128 + t] : -3.0e38f;
  if (t < 128) red[t] = v;
  __syncthreads();
  for (int s = 64; s > 0; s >>= 1) { if (t < s) red[t] = fmaxf(red[t], red[t + s]); __syncthreads(); }
  const float mx = red[0]; __syncthreads();
  const float e = (t < 128) ? expf(v - mx) : 0.0f;
  if (t < 128) red[t] = e;
  __syncthreads();
  for (int s = 64; s > 0; s >>= 1) { if (t < s) red[t] += red[t + s]; __syncthreads(); }
  const float inv = 1.0f / red[0];
  if (t < 128) {
    sw[t] = e * inv;
    out[2048 + b * 128 + t] = sw[t];
  }
  __syncthreads();
  float acc = 0.0f;
  for (int s = 0; s < 128; ++s)
    acc += sw[s] * x[((size_t)(b * 128 + s)) * 256 + t];
  out[b * 256 + t] = acc;
}

// ---------------------------------------------------------------------------
// Orchestration
// ---------------------------------------------------------------------------
extern "C" void kernel_launch(void* const* d_in, const int* in_sizes, int n_in,
                              void* d_out, int out_size, void* d_ws, size_t ws_size,
                              hipStream_t stream) {
  (void)in_sizes; (void)n_in; (void)out_size; (void)ws_size;
  const float* X   = (const float*)d_in[0];   // [8,1024,256]
  const int*   sb  = (const int*)  d_in[1];   // [8,128,2]
  const float* wq  = (const float*)d_in[2];
  const float* bq  = (const float*)d_in[3];
  const float* wk  = (const float*)d_in[4];
  const float* bk  = (const float*)d_in[5];
  const float* wv  = (const float*)d_in[6];
  const float* bv  = (const float*)d_in[7];
  const float* wo  = (const float*)d_in[8];
  const float* bo  = (const float*)d_in[9];
  const float* lng = (const float*)d_in[10];
  const float* lnb = (const float*)d_in[11];
  const float* w2s = (const float*)d_in[12];
  const float* b2s = (const float*)d_in[13];
  const float* lw[2][2][4];   // lstm[layer][fwd|bwd]{wih,whh,bih,bhh}
  for (int l = 0; l < 2; ++l)
    for (int d = 0; d < 2; ++d)
      for (int p = 0; p < 4; ++p)
        lw[l][d][p] = (const float*)d_in[14 + l * 8 + d * 4 + p];
  const float* a1w = (const float*)d_in[30];
  const float* a1b = (const float*)d_in[31];
  const float* a2w = (const float*)d_in[32];
  const float* a2b = (const float*)d_in[33];

  float* out  = (float*)d_out;
  float* attn = out + 3072;                  // [64,1024,1024] lives in d_out

  float* ws = (float*)d_ws;
  const size_t BW = (size_t)8 * 1024 * 256;  // 2,097,152 floats
  float* q   = ws;
  float* k   = ws + BW;
  float* v   = ws + 2 * BW;
  float* ctx = ws + 3 * BW;
  float* wa  = ws + 4 * BW;                  // out+residual, then LN in place
  // sentence-level scratch reuses q/k slots after attention is done
  float* pooled = ws;                         // [1024,256]
  float* xA     = pooled + 262144;            // [1024,256]
  float* xB     = xA + 262144;                // [1024,256]
  float* xW     = xB + 262144;                // [2][1024,512]
  float* sc     = xW + 1048576;               // [1024]

  // --- word-level Q/K/V projections (plain [B*W, 256] layout) ---
  gemm_wmma_f32<<<dim3(128, 16), dim3(128), 0, stream>>>(X, 256, wq, bq, nullptr, nullptr, q, 256);
  gemm_wmma_f32<<<dim3(128, 16), dim3(128), 0, stream>>>(X, 256, wk, bk, nullptr, nullptr, k, 256);
  gemm_wmma_f32<<<dim3(128, 16), dim3(128), 0, stream>>>(X, 256, wv, bv, nullptr, nullptr, v, 256);

  // --- scores -> softmax (in d_out) -> ctx ---
  attn_scores_wmma<<<dim3(64, 16, 64), dim3(32), 0, stream>>>(q, k, attn, 0.17677669529663687f);
  softmax1024<<<dim3(65536), dim3(256), 0, stream>>>(attn);
  attn_ctx_wmma<<<dim3(64, 64), dim3(32), 0, stream>>>(attn, v, ctx);

  // --- output projection + residual, then LayerNorm in place ---
  gemm_wmma_f32<<<dim3(128, 16), dim3(128), 0, stream>>>(ctx, 256, wo, bo, nullptr, X, wa, 256);
  layernorm256<<<dim3(8192), dim3(256), 0, stream>>>(wa, lng, lnb);

  // --- ragged mean pooling + sentence projection ---
  pool_mean<<<dim3(1024), dim3(256), 0, stream>>>(wa, sb, pooled);
  gemm_wmma_f32<<<dim3(16, 16), dim3(128), 0, stream>>>(pooled, 256, w2s, b2s, nullptr, nullptr, xA, 256);

  // --- 2-layer bidirectional LSTM ---
  for (int l = 0; l < 2; ++l) {
    const float* xin = (l == 0) ? xA : xB;
    float* xout      = (l == 0) ? xB : xA;
    gemm_wmma_f32<<<dim3(16, 32), dim3(128), 0, stream>>>(xin, 256, lw[l][0][0],
        lw[l][0][2], lw[l][0][3], nullptr, xW, 512);
    gemm_wmma_f32<<<dim3(16, 32), dim3(128), 0, stream>>>(xin, 256, lw[l][1][0],
        lw[l][1][2], lw[l][1][3], nullptr, xW + 524288, 512);
    lstm_rec<<<dim3(2), dim3(512), 0, stream>>>(xW, lw[l][0][1], lw[l][1][1], xout);
  }

  // --- sentence attention -> doc vector (final x is in xA) ---
  sent_score<<<dim3(1024), dim3(128), 0, stream>>>(xA, a1w, a1b, a2w, a2b, sc);
  doc_reduce<<<dim3(8), dim3(256), 0, stream>>>(sc, xA, out);
}